// Block_2585570312247
// MI455X (gfx1250) — compile-verified
//
#include <hip/hip_runtime.h>

typedef __attribute__((ext_vector_type(2))) float v2f;
typedef __attribute__((ext_vector_type(8))) float v8f;

constexpr int NB   = 4;
constexpr int CIN  = 128;
constexpr int COUT = 256;
constexpr int NPTS = 4096;
constexpr int KNN  = 16;

// LDS row strides (padded to avoid bank conflicts; keep float2 8B alignment)
constexpr int LDA = 66;
constexpr int LDB = 66;

// ---------------------------------------------------------------------------
// WMMA GEMM (fp32, 16x16x4): Y[b][o][n] = sum_r A[o][r] * Bmat[b][r][n] + bias[o]
//   GATHER=false: Bmat[r][n] = X[b][r][n]                       (residual 1x1 conv)
//   GATHER=true : Bmat[c*16+k][n] = X[b][c][ idx[b][n][k] ]     (point conv)
// Block: 256 threads = 8 waves; tile = 128(o) x 16(n); K staged 64 at a time.
// ---------------------------------------------------------------------------
template <bool GATHER>
__global__ __launch_bounds__(256) void wmma_gemm_k(
    const float* __restrict__ A, const float* __restrict__ X,
    const int* __restrict__ idx, const float* __restrict__ bias,
    float* __restrict__ Y, int Kdepth, int astride, int xrows)
{
    __shared__ float sA[128 * LDA];   // A tile: 128 rows x 64 cols
    __shared__ float sBt[16 * LDB];   // B tile transposed: [n][r]
    __shared__ int   sIdx[16 * KNN];

    const int t    = threadIdx.x;
    const int lane = t & 31;
    const int w    = t >> 5;          // wave id 0..7 -> o sub-tile
    const int lm   = lane & 15;       // M for A-frag, N for B-frag, N for C/D
    const int lh   = lane >> 4;       // lane-half
    const int n0   = blockIdx.x * 16;
    const int o0   = blockIdx.y * 128;
    const int b    = blockIdx.z;

    if (GATHER) {
        // 16 n-values x 16 k -> 256 ints, one per thread
        const int nl = t >> 4, k = t & 15;
        sIdx[t] = idx[((size_t)(b * NPTS + n0 + nl)) * KNN + k];
    }
    __syncthreads();

    v8f acc = {};

    for (int r0 = 0; r0 < Kdepth; r0 += 64) {
        // ---- stage A: 128x64, coalesced rows of W ----
        #pragma unroll
        for (int i = 0; i < 32; ++i) {
            const int e = t + 256 * i;
            const int row = e >> 6, col = e & 63;
            sA[row * LDA + col] = A[(size_t)(o0 + row) * astride + r0 + col];
        }
        if (r0 + 64 < Kdepth) {
            // prefetch next A chunk (global_prefetch_b8)
            __builtin_prefetch(A + (size_t)(o0 + (t >> 1)) * astride + r0 + 64, 0, 1);
        }
        // ---- stage B: 64(r) x 16(n), stored transposed as sBt[n][r] ----
        #pragma unroll
        for (int i = 0; i < 4; ++i) {
            const int e  = t + 256 * i;
            const int nl = e & 15, rl = e >> 4;
            float v;
            if (GATHER) {
                const int r = r0 + rl;
                const int c = r >> 4, k = r & 15;
                const int col = sIdx[nl * KNN + k];
                v = X[((size_t)(b * xrows + c)) * NPTS + col];
            } else {
                v = X[((size_t)(b * xrows + r0 + rl)) * NPTS + n0 + nl];
            }
            sBt[nl * LDB + rl] = v;
        }
        __syncthreads();

        // ---- 16 WMMA f32 16x16x4 per wave over this 64-deep chunk ----
        const float* ar = &sA[(16 * w + lm) * LDA + 2 * lh];
        const float* br = &sBt[lm * LDB + 2 * lh];
        #pragma unroll
        for (int kk = 0; kk < 16; ++kk) {
            v2f av = *(const v2f*)(ar + 4 * kk);  // A[M=lm][K=4kk+2lh+{0,1}]
            v2f bv = *(const v2f*)(br + 4 * kk);  // B[K=4kk+2lh+{0,1}][N=lm]
            acc = __builtin_amdgcn_wmma_f32_16x16x4_f32(
                false, av, false, bv, (short)0, acc, false, false);
        }
        __syncthreads();
    }

    // C/D layout: VGPR rr -> M = rr + 8*lh, N = lm
    #pragma unroll
    for (int rr = 0; rr < 8; ++rr) {
        const int o = o0 + 16 * w + rr + 8 * lh;
        Y[((size_t)(b * COUT + o)) * NPTS + n0 + lm] = acc[rr] + bias[o];
    }
}

// ---------------------------------------------------------------------------
// Row statistics for InstanceNorm: mean & rsqrt(var+eps) per (b, channel) row
// ---------------------------------------------------------------------------
__global__ __launch_bounds__(256) void row_stats_k(
    const float* __restrict__ Y, float* __restrict__ mean, float* __restrict__ rstd)
{
    const int row = blockIdx.x;
    const int t   = threadIdx.x;
    const float* p = Y + (size_t)row * NPTS;
    float s = 0.f, s2 = 0.f;
    #pragma unroll
    for (int j = 0; j < 16; ++j) {
        float v = p[t + 256 * j];
        s += v;
        s2 = fmaf(v, v, s2);
    }
    __shared__ float a1[256], a2[256];
    a1[t] = s; a2[t] = s2;
    __syncthreads();
    for (int st = 128; st > 0; st >>= 1) {
        if (t < st) { a1[t] += a1[t + st]; a2[t] += a2[t + st]; }
        __syncthreads();
    }
    if (t == 0) {
        float m   = a1[0] * (1.0f / NPTS);
        float var = a2[0] * (1.0f / NPTS) - m * m;
        mean[row] = m;
        rstd[row] = rsqrtf(var + 1e-5f);
    }
}

// ---------------------------------------------------------------------------
// y = (x - m) * rstd * g + be  [+ relu]; float4 per thread
// ---------------------------------------------------------------------------
__global__ __launch_bounds__(256) void apply_norm_k(
    const float* __restrict__ Yin, const float* __restrict__ mean,
    const float* __restrict__ rstd, const float* __restrict__ g,
    const float* __restrict__ be, float* __restrict__ Yout, int relu)
{
    const int i    = blockIdx.x * 256 + threadIdx.x;
    const int row  = (i * 4) >> 12;          // /4096
    const int o    = row & (COUT - 1);
    const float m  = mean[row], r = rstd[row];
    const float gg = g[o], bb = be[o];
    float4 v = ((const float4*)Yin)[i];
    float4 u;
    u.x = (v.x - m) * r * gg + bb;
    u.y = (v.y - m) * r * gg + bb;
    u.z = (v.z - m) * r * gg + bb;
    u.w = (v.w - m) * r * gg + bb;
    if (relu) {
        u.x = fmaxf(u.x, 0.f); u.y = fmaxf(u.y, 0.f);
        u.z = fmaxf(u.z, 0.f); u.w = fmaxf(u.w, 0.f);
    }
    ((float4*)Yout)[i] = u;
}

// out = relu(x_in + (y - m) * rstd * g + be)
__global__ __launch_bounds__(256) void final_k(
    const float* __restrict__ Xin, const float* __restrict__ Yin,
    const float* __restrict__ mean, const float* __restrict__ rstd,
    const float* __restrict__ g, const float* __restrict__ be,
    float* __restrict__ Out)
{
    const int i   = blockIdx.x * 256 + threadIdx.x;
    const int row = (i * 4) >> 12;
    const int o   = row & (COUT - 1);
    const float m = mean[row], r = rstd[row];
    const float gg = g[o], bb = be[o];
    float4 v = ((const float4*)Yin)[i];
    float4 x = ((const float4*)Xin)[i];
    float4 u;
    u.x = fmaxf(x.x + (v.x - m) * r * gg + bb, 0.f);
    u.y = fmaxf(x.y + (v.y - m) * r * gg + bb, 0.f);
    u.z = fmaxf(x.z + (v.z - m) * r * gg + bb, 0.f);
    u.w = fmaxf(x.w + (v.w - m) * r * gg + bb, 0.f);
    ((float4*)Out)[i] = u;
}

// ---------------------------------------------------------------------------
// kNN: per batch, all points + |p|^2 staged in LDS (64 KB); one query/thread,
// fully-unrolled 16-entry sorted insertion (stable ties: smaller index first,
// matching jax.lax.top_k). d = |q|^2 + |p|^2 - 2 q.p (reference formula).
// ---------------------------------------------------------------------------
__global__ __launch_bounds__(256) void knn_k(
    const float* __restrict__ pts, int* __restrict__ idxOut)
{
    __shared__ float spx[NPTS], spy[NPTS], spz[NPTS], ssq[NPTS];
    const int t = threadIdx.x;
    const int b = blockIdx.y;
    const int n = blockIdx.x * 256 + t;
    const float* pb = pts + (size_t)b * 3 * NPTS;
    #pragma unroll
    for (int i = 0; i < 16; ++i) {
        const int m = t + 256 * i;
        float x = pb[m], y = pb[NPTS + m], z = pb[2 * NPTS + m];
        spx[m] = x; spy[m] = y; spz[m] = z;
        ssq[m] = x * x + y * y + z * z;
    }
    __syncthreads();

    const float qx = spx[n], qy = spy[n], qz = spz[n], qs = ssq[n];
    float bd[16];
    int   bi[16];
    #pragma unroll
    for (int j = 0; j < 16; ++j) { bd[j] = 3.4e38f; bi[j] = 0; }

    for (int m = 0; m < NPTS; ++m) {
        const float d =
            qs + ssq[m] - 2.0f * (qx * spx[m] + qy * spy[m] + qz * spz[m]);
        if (d < bd[15]) {
            int rank = 0;
            #pragma unroll
            for (int j = 0; j < 16; ++j) rank += (bd[j] <= d) ? 1 : 0;
            #pragma unroll
            for (int j = 15; j > 0; --j)
                if (j > rank) { bd[j] = bd[j - 1]; bi[j] = bi[j - 1]; }
            #pragma unroll
            for (int j = 0; j < 16; ++j)
                if (j == rank) { bd[j] = d; bi[j] = m; }
        }
    }
    int* op = idxOut + ((size_t)(b * NPTS + n)) * KNN;
    #pragma unroll
    for (int j = 0; j < 16; ++j) op[j] = bi[j];
}

// ---------------------------------------------------------------------------
extern "C" void kernel_launch(void* const* d_in, const int* in_sizes, int n_in,
                              void* d_out, int out_size, void* d_ws, size_t ws_size,
                              hipStream_t stream)
{
    const float* input  = (const float*)d_in[0];
    const float* points = (const float*)d_in[1];
    const float* W_res  = (const float*)d_in[2];
    const float* b_res  = (const float*)d_in[3];
    const float* g_res  = (const float*)d_in[4];
    const float* be_res = (const float*)d_in[5];
    const float* W1     = (const float*)d_in[6];
    const float* b1     = (const float*)d_in[7];
    const float* g1     = (const float*)d_in[8];
    const float* be1    = (const float*)d_in[9];
    const float* W2     = (const float*)d_in[10];
    const float* b2     = (const float*)d_in[11];
    const float* g2     = (const float*)d_in[12];
    const float* be2    = (const float*)d_in[13];

    // d_out layout: out [4,256,4096] f32 | points [4,3,4096] f32 | idx [4,4096,16] i32
    float* out        = (float*)d_out;
    float* points_out = out + (size_t)NB * COUT * NPTS;
    int*   idx_out    = (int*)(points_out + (size_t)NB * 3 * NPTS);

    // workspace
    constexpr size_t TSZ = (size_t)NB * COUT * NPTS;  // 4,194,304 floats
    float* x_in = (float*)d_ws;
    float* buf1 = x_in + TSZ;
    float* buf2 = buf1 + TSZ;
    float* mean = buf2 + TSZ;
    float* rstd = mean + 2048;

    const dim3 gblk(256);
    const dim3 ggrid(NPTS / 16, COUT / 128, NB);
    const dim3 eblk(256);
    const dim3 egrid((unsigned)(TSZ / 4 / 256));      // float4 elementwise

    // 1) residual 1x1 conv (raw) -> buf1
    wmma_gemm_k<false><<<ggrid, gblk, 0, stream>>>(W_res, input, nullptr, b_res,
                                                   buf1, CIN, CIN, CIN);
    // 2) instance norm -> x_in
    row_stats_k<<<dim3(NB * COUT), dim3(256), 0, stream>>>(buf1, mean, rstd);
    apply_norm_k<<<egrid, eblk, 0, stream>>>(buf1, mean, rstd, g_res, be_res, x_in, 0);

    // 3) kNN -> idx_out (also an output), points passthrough
    knn_k<<<dim3(NPTS / 256, NB), dim3(256), 0, stream>>>(points, idx_out);
    hipMemcpyAsync(points_out, points, sizeof(float) * NB * 3 * NPTS,
                   hipMemcpyDeviceToDevice, stream);

    // 4) enc1: gathered point conv -> buf1 ; IN + relu in place
    wmma_gemm_k<true><<<ggrid, gblk, 0, stream>>>(W1, x_in, idx_out, b1,
                                                  buf1, COUT * KNN, COUT * KNN, COUT);
    row_stats_k<<<dim3(NB * COUT), dim3(256), 0, stream>>>(buf1, mean, rstd);
    apply_norm_k<<<egrid, eblk, 0, stream>>>(buf1, mean, rstd, g1, be1, buf1, 1);

    // 5) enc2: gathered point conv -> buf2 ; IN ; residual add + relu -> out
    wmma_gemm_k<true><<<ggrid, gblk, 0, stream>>>(W2, buf1, idx_out, b2,
                                                  buf2, COUT * KNN, COUT * KNN, COUT);
    row_stats_k<<<dim3(NB * COUT), dim3(256), 0, stream>>>(buf2, mean, rstd);
    final_k<<<egrid, eblk, 0, stream>>>(x_in, buf2, mean, rstd, g2, be2, out);
}